// MACEForce_6983616824052
// MI455X (gfx1250) — compile-verified
//
#include <hip/hip_runtime.h>

typedef __attribute__((ext_vector_type(2))) float v2f;
typedef __attribute__((ext_vector_type(8))) float v8f;

#define NB     8
#define HID    64
#define DCH    32
#define NELEM  10
#define PI_F   3.14159265358979f
#define E2KJ   96.4853f
#define NM2ANG 10.0f
#define INV_RMAX 0.2f               // 1/5.0
#define BESSEL_PREF 0.632455532034f // sqrt(2/5)

// fast hardware reciprocal (v_rcp_f32, ~1 ulp)
__device__ __forceinline__ float fast_rcp(float x) {
  return __builtin_amdgcn_rcpf(x);
}

// silu(x) = x * sigmoid(x). CDNA5 has a hardware v_tanh_f32, so use the exact
// identity x*sigmoid(x) = (x/2)*(1 + tanh(x/2)): 1 TRANS op instead of the
// exp+rcp pair -- halves the transcendental-pipe load, which is the kernel's
// arithmetic ceiling (102M silu evals vs only ~77MB of HBM traffic).
__device__ __forceinline__ float fast_silu(float x) {
#if __has_builtin(__builtin_amdgcn_tanhf)
  float h = 0.5f * x;
  return h + h * __builtin_amdgcn_tanhf(h);
#else
  return x * fast_rcp(1.f + __expf(-x));
#endif
}

// ---------------------------------------------------------------------------
// u[z,k] = sum_c W2[k, c] * W_out[c] * W_z[z, c]   (c over first D=32 cols)
// Only the l=0 slice of W2 survives the scalar-energy readout.
// ---------------------------------------------------------------------------
__global__ void mace_prep_u(const float* __restrict__ Wz,
                            const float* __restrict__ W2,
                            const float* __restrict__ Wout,
                            float* __restrict__ u) {
  int t = blockIdx.x * blockDim.x + threadIdx.x;
  if (t >= NELEM * HID) return;
  int z = t / HID, k = t % HID;
  float s = 0.f;
#pragma unroll
  for (int c = 0; c < DCH; ++c)
    s += W2[k * (4 * DCH) + c] * Wout[c] * Wz[z * DCH + c];
  u[t] = s;
}

// species index from one-hot node_attrs
__global__ void mace_prep_spec(const float* __restrict__ attrs,
                               int* __restrict__ spec, int N) {
  int i = blockIdx.x * blockDim.x + threadIdx.x;
  if (i >= N) return;
  int s = 0;
#pragma unroll
  for (int z = 0; z < NELEM; ++z)
    if (attrs[i * NELEM + z] > 0.5f) s = z;
  spec[i] = s;
}

__global__ void mace_zero(float* out) { out[0] = 0.f; }

// ---------------------------------------------------------------------------
// Per-edge: radial basis -> WMMA (16 edges x 64 hid, K=8) -> silu -> weight by
// u[species(snd)] -> single global scalar reduction (order inside the sum is
// free, so no per-edge cross-lane reduce is needed).
// ---------------------------------------------------------------------------
__global__ __launch_bounds__(256) void mace_edge(
    const float* __restrict__ pos, const float* __restrict__ shifts,
    const int* __restrict__ eidx, const float* __restrict__ W1,
    const float* __restrict__ b1, const float* __restrict__ u_ws,
    const int* __restrict__ spec_ws, float* __restrict__ out, int E) {
  // u table gets an extra all-zero row (index NELEM) so masked edges need no
  // divergent guard: their spec points at the zero row.
  __shared__ float u_lds[(NELEM + 1) * HID];
  __shared__ float basis_lds[256 * 9]; // stride 9: bank-conflict padding
  __shared__ alignas(16) int spec_lds[256];
  __shared__ float red[8];

  const int tid  = threadIdx.x;
  const int lane = tid & 31;
  const int wid  = tid >> 5;
  const int ln   = lane & 15;   // position within 16-lane half
  const int hi   = lane >> 4;   // which half (K-split in f32 WMMA layouts)

  for (int i = tid; i < (NELEM + 1) * HID; i += 256)
    u_lds[i] = (i < NELEM * HID) ? u_ws[i] : 0.f;

  // Wave-uniform B operands: W1 (8x64) as 4x16 K-chunks in f32 B layout
  // (VGPR0: lanes0-15 K=+0, lanes16-31 K=+2; VGPR1: K=+1 / K=+3)
  v2f B[2][4];
  float b1k[4];
#pragma unroll
  for (int q = 0; q < 2; ++q)
#pragma unroll
    for (int j = 0; j < 4; ++j) {
      B[q][j].x = W1[(4 * q + 2 * hi + 0) * HID + j * 16 + ln];
      B[q][j].y = W1[(4 * q + 2 * hi + 1) * HID + j * 16 + ln];
    }
#pragma unroll
  for (int j = 0; j < 4; ++j) b1k[j] = b1[j * 16 + ln];

  // --- per-lane edge: Bessel basis x poly cutoff -> LDS ---
  const int e = blockIdx.x * 256 + tid;
  if (e < E) {
    int snd = eidx[e];
    int rcv = eidx[E + e];
    float dx = (pos[3 * rcv + 0] - pos[3 * snd + 0]) * NM2ANG + shifts[3 * e + 0];
    float dy = (pos[3 * rcv + 1] - pos[3 * snd + 1]) * NM2ANG + shifts[3 * e + 1];
    float dz = (pos[3 * rcv + 2] - pos[3 * snd + 2]) * NM2ANG + shifts[3 * e + 2];
    float r = __builtin_amdgcn_sqrtf(dx * dx + dy * dy + dz * dz + 1e-12f);
    float uu = r * INV_RMAX;
    float u2 = uu * uu;
    float u6 = u2 * u2 * u2;
    float fcp = 1.f + u6 * (-28.f + uu * (48.f - 21.f * uu)); // p=6 cutoff
    float fc = (uu < 1.f) ? fcp : 0.f;
    // sin(n*theta) via Chebyshev recurrence: 1 sincos instead of 8 sins
    float theta = PI_F * uu;
    float s1, c1;
    __sincosf(theta, &s1, &c1);
    float scale = BESSEL_PREF * fc * fast_rcp(r);
    float twoc = 2.f * c1, sm2 = 0.f, sm1 = s1;
    basis_lds[tid * 9 + 0] = scale * s1;
#pragma unroll
    for (int n = 1; n < NB; ++n) {
      float sn = twoc * sm1 - sm2;
      sm2 = sm1; sm1 = sn;
      basis_lds[tid * 9 + n] = scale * sn;
    }
    spec_lds[tid] = spec_ws[snd];
  } else {
#pragma unroll
    for (int n = 0; n < NB; ++n) basis_lds[tid * 9 + n] = 0.f;
    spec_lds[tid] = NELEM; // zero row of u table -> zero contribution
  }
  __syncthreads();

  // --- two 16-edge WMMA tiles per wave ---
  float acc = 0.f;
#pragma unroll
  for (int t = 0; t < 2; ++t) {
    int rowL = wid * 32 + t * 16 + ln; // edge row in LDS (A layout: M=ln)
    v2f a0, a1;
    a0.x = basis_lds[rowL * 9 + 2 * hi + 0];
    a0.y = basis_lds[rowL * 9 + 2 * hi + 1];
    a1.x = basis_lds[rowL * 9 + 4 + 2 * hi + 0];
    a1.y = basis_lds[rowL * 9 + 4 + 2 * hi + 1];

    // hidden[16x64] = basis[:,0:4]@W1[0:4,:] + basis[:,4:8]@W1[4:8,:]
    v8f c[4];
#pragma unroll
    for (int j = 0; j < 4; ++j) {
      v8f z = {0.f, 0.f, 0.f, 0.f, 0.f, 0.f, 0.f, 0.f};
      z = __builtin_amdgcn_wmma_f32_16x16x4_f32(false, a0, false, B[0][j],
                                                (short)0, z, false, false);
      c[j] = __builtin_amdgcn_wmma_f32_16x16x4_f32(false, a1, false, B[1][j],
                                                   (short)0, z, false, false);
    }

    // C layout: lane l, vgpr v -> edge M = v + 8*hi, k = j*16 + ln.
    // The 8 edge species this lane touches are consecutive ints in LDS.
    const int spBase = wid * 32 + t * 16 + 8 * hi;
    const int4 spA = *(const int4*)&spec_lds[spBase];
    const int4 spB = *(const int4*)&spec_lds[spBase + 4];
    int sp[8] = {spA.x, spA.y, spA.z, spA.w, spB.x, spB.y, spB.z, spB.w};

#pragma unroll
    for (int v = 0; v < 8; ++v) {
      const float* urow = &u_lds[sp[v] * HID + ln]; // stride-16 pair loads
      float g0 = urow[0], g1 = urow[16], g2 = urow[32], g3 = urow[48];
      acc += fast_silu(c[0][v] + b1k[0]) * g0;
      acc += fast_silu(c[1][v] + b1k[1]) * g1;
      acc += fast_silu(c[2][v] + b1k[2]) * g2;
      acc += fast_silu(c[3][v] + b1k[3]) * g3;
    }
  }

  // single wave reduce, then block reduce, one atomic per block
  acc += __shfl_xor(acc, 16);
  acc += __shfl_xor(acc, 8);
  acc += __shfl_xor(acc, 4);
  acc += __shfl_xor(acc, 2);
  acc += __shfl_xor(acc, 1);
  if (lane == 0) red[wid] = acc;
  __syncthreads();
  if (tid == 0) {
    float tot = 0.f;
#pragma unroll
    for (int w = 0; w < 8; ++w) tot += red[w];
    atomicAdd(out, tot * E2KJ);
  }
}

// ---------------------------------------------------------------------------
extern "C" void kernel_launch(void* const* d_in, const int* in_sizes, int n_in,
                              void* d_out, int out_size, void* d_ws,
                              size_t ws_size, hipStream_t stream) {
  const float* pos    = (const float*)d_in[0];
  const float* attrs  = (const float*)d_in[1];
  const float* shifts = (const float*)d_in[2];
  const float* Wz     = (const float*)d_in[3];
  const float* W1     = (const float*)d_in[4];
  const float* b1     = (const float*)d_in[5];
  const float* W2     = (const float*)d_in[6];
  const float* Wout   = (const float*)d_in[7];
  const int*   eidx   = (const int*)d_in[8];
  float* out = (float*)d_out;

  const int N = in_sizes[0] / 3;
  const int E = in_sizes[8] / 2;

  float* u_ws    = (float*)d_ws;                // 2560 B
  int*   spec_ws = (int*)((char*)d_ws + 4096);  // N ints

  mace_zero<<<1, 1, 0, stream>>>(out);
  mace_prep_u<<<(NELEM * HID + 255) / 256, 256, 0, stream>>>(Wz, W2, Wout, u_ws);
  mace_prep_spec<<<(N + 255) / 256, 256, 0, stream>>>(attrs, spec_ws, N);
  mace_edge<<<(E + 255) / 256, 256, 0, stream>>>(pos, shifts, eidx, W1, b1,
                                                 u_ws, spec_ws, out, E);
}